// MANO_SMPL_6863357739234
// MI455X (gfx1250) — compile-verified
//
#include <hip/hip_runtime.h>

// ---------------------------------------------------------------------------
// MANO/SMPL forward pass for MI455X (gfx1250, wave32).
// Dominant GEMM (pose_feat[2048x135] @ posedirs[135x2334]) runs on
// V_WMMA_F32_16X16X4_F32 with fp32 accumulation; everything else is
// latency-trivial VALU work.
// ---------------------------------------------------------------------------

#define B_    2048
#define V_    778
#define NJ_   16
#define VC    2334          // V*3
#define VCP   2336          // padded to multiple of 16
#define KP    136           // 135 pose-feat dims padded to multiple of 4
#define SCALE_ 1000.0f

typedef __attribute__((ext_vector_type(2))) float v2f;
typedef __attribute__((ext_vector_type(8))) float v8f;

__constant__ int c_parents[16] = {-1,0,1,2,0,4,5,0,7,8,0,10,11,0,13,14};

// ---- 1) zero-pad posedirs (135 x 2334) -> (136 x 2336) ---------------------
__global__ __launch_bounds__(256) void k_pad_posedirs(
    const float* __restrict__ pd, float* __restrict__ pdp)
{
    int idx = blockIdx.x * 256 + threadIdx.x;
    if (idx >= KP * VCP) return;
    int r = idx / VCP, cc = idx - r * VCP;
    float v = 0.0f;
    if (r < 135 && cc < VC) v = pd[(size_t)r * VC + cc];
    pdp[idx] = v;
}

// ---- 2) v_shaped = beta @ shapedirs + v_template ---------------------------
__global__ __launch_bounds__(256) void k_shape(
    const float* __restrict__ beta,       // [B,10]
    const float* __restrict__ shapedirs,  // [10,2334]
    const float* __restrict__ v_template, // [2334]
    float* __restrict__ vs)               // [B,2336] (cols 2334/2335 zero)
{
    const int b   = blockIdx.y;
    const int col = blockIdx.x * 256 + threadIdx.x;
    if (col >= VCP) return;
    float acc = 0.0f;
    if (col < VC) {
        acc = v_template[col];
        const float* bb = beta + (size_t)b * 10;
#pragma unroll
        for (int k = 0; k < 10; ++k)
            acc += bb[k] * shapedirs[(size_t)k * VC + col];
    }
    vs[(size_t)b * VCP + col] = acc;
}

// ---- 3) pose PCA + Rodrigues -> Rs [B,16,9], pose_feat_pad [B,136] ---------
__global__ __launch_bounds__(64) void k_pose(
    const float* __restrict__ theta,       // [B,48]
    const float* __restrict__ hands_comp,  // [45,45]
    const float* __restrict__ hands_mean,  // [45]
    float* __restrict__ Rs,                // [B,16,9]
    float* __restrict__ pf)                // [B,136]
{
    __shared__ float fp[48];
    const int b = blockIdx.x;
    const int t = threadIdx.x;
    const float* th = theta + (size_t)b * 48;
    if (t < 45) {
        float acc = hands_mean[t];
        for (int q = 0; q < 45; ++q)
            acc += th[3 + q] * hands_comp[q * 45 + t];
        fp[3 + t] = acc;
    }
    if (t < 3) fp[t] = th[t];
    __syncthreads();
    if (t < 16) {
        float rx = fp[t*3+0], ry = fp[t*3+1], rz = fp[t*3+2];
        float ang = sqrtf(rx*rx + ry*ry + rz*rz + 1e-8f);
        float inv = 1.0f / ang;
        float ax = rx*inv, ay = ry*inv, az = rz*inv;
        float c = cosf(ang), s = sinf(ang), ic = 1.0f - c;
        float R[9];
        R[0] = c + ic*ax*ax;    R[1] = ic*ax*ay - s*az; R[2] = ic*ax*az + s*ay;
        R[3] = ic*ay*ax + s*az; R[4] = c + ic*ay*ay;    R[5] = ic*ay*az - s*ax;
        R[6] = ic*az*ax - s*ay; R[7] = ic*az*ay + s*ax; R[8] = c + ic*az*az;
        float* rp = Rs + ((size_t)b * 16 + t) * 9;
#pragma unroll
        for (int e = 0; e < 9; ++e) rp[e] = R[e];
        if (t >= 1) {
            float* pp = pf + (size_t)b * KP + (t - 1) * 9;
#pragma unroll
            for (int e = 0; e < 9; ++e)
                pp[e] = R[e] - ((e == 0 || e == 4 || e == 8) ? 1.0f : 0.0f);
        } else {
            pf[(size_t)b * KP + 135] = 0.0f;  // K pad
        }
    }
}

// ---- 4) J[b,j,d] = sum_v v_shaped[b,v,d] * Jreg[v,j] -----------------------
__global__ __launch_bounds__(64) void k_jreg(
    const float* __restrict__ vs,    // [B,2336]
    const float* __restrict__ Jreg,  // [778,21] (use cols 0..15)
    float* __restrict__ J)           // [B,48]
{
    const int b = blockIdx.x;
    const int t = threadIdx.x;
    if (t >= 48) return;
    const int j = t / 3, d = t - j * 3;
    const float* vb = vs + (size_t)b * VCP;
    float acc = 0.0f;
    for (int v = 0; v < V_; ++v)
        acc += vb[v * 3 + d] * Jreg[v * 21 + j];
    J[(size_t)b * 48 + t] = acc;
}

// ---- 5) kinematic chain: A [B,16,12], joints[0..15] ------------------------
__global__ __launch_bounds__(32) void k_kinematics(
    const float* __restrict__ Rs,   // [B,16,9]
    const float* __restrict__ J,    // [B,16,3]
    float* __restrict__ Amat,       // [B,16,12] 3x4 row-major
    float* __restrict__ joints)     // [B,21,3]
{
    __shared__ float Gs[16 * 12];
    const int b = blockIdx.x;
    const int j = threadIdx.x;
    float R[9] = {0}, t[3] = {0}, Jj[3] = {0};
    int depth = 0, par = 0;
    if (j < 16) {
        par   = c_parents[j];
        depth = (j == 0) ? 0 : ((j - 1) % 3) + 1;
        const float* rp = Rs + ((size_t)b * 16 + j) * 9;
#pragma unroll
        for (int e = 0; e < 9; ++e) R[e] = rp[e];
        const float* jp = J + ((size_t)b * 16 + j) * 3;
#pragma unroll
        for (int d = 0; d < 3; ++d) Jj[d] = jp[d];
        if (j == 0) {
            t[0] = Jj[0]; t[1] = Jj[1]; t[2] = Jj[2];
        } else {
            const float* pp = J + ((size_t)b * 16 + par) * 3;
#pragma unroll
            for (int d = 0; d < 3; ++d) t[d] = Jj[d] - pp[d];
        }
    }
    if (j == 0) {
        float* g = Gs;
        g[0]=R[0]; g[1]=R[1]; g[2] =R[2]; g[3] =t[0];
        g[4]=R[3]; g[5]=R[4]; g[6] =R[5]; g[7] =t[1];
        g[8]=R[6]; g[9]=R[7]; g[10]=R[8]; g[11]=t[2];
    }
    __syncthreads();
    for (int lev = 1; lev <= 3; ++lev) {
        if (j < 16 && depth == lev) {
            const float* gp = Gs + par * 12;
            float g[12];
#pragma unroll
            for (int m = 0; m < 3; ++m) {
#pragma unroll
                for (int n = 0; n < 3; ++n)
                    g[m*4+n] = gp[m*4+0]*R[0*3+n] + gp[m*4+1]*R[1*3+n] + gp[m*4+2]*R[2*3+n];
                g[m*4+3] = gp[m*4+0]*t[0] + gp[m*4+1]*t[1] + gp[m*4+2]*t[2] + gp[m*4+3];
            }
            float* gd = Gs + j * 12;
#pragma unroll
            for (int e = 0; e < 12; ++e) gd[e] = g[e];
        }
        __syncthreads();
    }
    if (j < 16) {
        const float* g = Gs + j * 12;
        float* ap = Amat + ((size_t)b * 16 + j) * 12;
        float* jo = joints + ((size_t)b * 21 + j) * 3;
#pragma unroll
        for (int m = 0; m < 3; ++m) {
            float tt = g[m*4+3] - (g[m*4+0]*Jj[0] + g[m*4+1]*Jj[1] + g[m*4+2]*Jj[2]);
            ap[m*4+0] = g[m*4+0]; ap[m*4+1] = g[m*4+1];
            ap[m*4+2] = g[m*4+2]; ap[m*4+3] = tt;
            jo[m] = g[m*4+3] * SCALE_;
        }
    }
}

// ---- 6) v_posed = v_shaped + pose_feat @ posedirs  (WMMA fp32) -------------
// One wave per 16x16 tile of [2048 x 2336]; K = 136 in steps of 4 via
// V_WMMA_F32_16X16X4_F32. In-place accumulate on the v_shaped buffer.
__global__ __launch_bounds__(32) void k_gemm_posedirs(
    const float* __restrict__ Apad,  // [2048 x 136]  pose_feat (zero-padded)
    const float* __restrict__ Bpad,  // [136 x 2336]  posedirs (zero-padded)
    float* __restrict__ Cbuf)        // [2048 x 2336] v_shaped -> v_posed
{
    const int lane = threadIdx.x;
    const int hf = lane >> 4;        // half-wave select
    const int l  = lane & 15;
    const int m0 = blockIdx.y << 4;
    const int n0 = blockIdx.x << 4;

    // C/D layout: VGPR r -> row m0 + r + hf*8, col n0 + l
    float* crow = Cbuf + (size_t)(m0 + hf * 8) * VCP + n0 + l;
    v8f c;
#pragma unroll
    for (int r = 0; r < 8; ++r) c[r] = crow[(size_t)r * VCP];

    // A 16x4 fp32: lanes 0-15 hold {K0,K1}, lanes 16-31 hold {K2,K3}, M = l
    const v2f*   ap = (const v2f*)(Apad + (size_t)(m0 + l) * KP + hf * 2);
    // B 4x16 fp32: v0 = row (k + 2*hf), v1 = row (k + 2*hf + 1), col = n0+l
    const float* bp = Bpad + (size_t)(hf * 2) * VCP + n0 + l;

#pragma unroll 2
    for (int k = 0; k < KP; k += 4) {
        v2f a = *ap;
        v2f bv; bv.x = bp[0]; bv.y = bp[VCP];
        c = __builtin_amdgcn_wmma_f32_16x16x4_f32(
                false, a, false, bv, (short)0, c, false, false);
        ap += 2;             // advance 4 floats
        bp += 4 * VCP;       // advance 4 rows
    }

    if (n0 + l < VC) {
#pragma unroll
        for (int r = 0; r < 8; ++r) crow[(size_t)r * VCP] = c[r];
    } else {
#pragma unroll
        for (int r = 0; r < 8; ++r) crow[(size_t)r * VCP] = 0.0f;
    }
}

// ---- 7) skinning: T = sum_j w[v,j]*A[b,j]; verts = T * [v_posed,1] ---------
__global__ __launch_bounds__(256) void k_skin(
    const float* __restrict__ vp,    // [B,2336] v_posed
    const float* __restrict__ Amat,  // [B,192]
    const float* __restrict__ w,     // [778,16]
    float* __restrict__ verts,       // [B,778,3]
    float* __restrict__ joints)      // [B,21,3]
{
    __shared__ float As[192];
    const int b = blockIdx.y;
    const int t = threadIdx.x;
    if (t < 192) As[t] = Amat[(size_t)b * 192 + t];
    __syncthreads();
    const int v = blockIdx.x * 256 + t;
    if (v >= V_) return;
    float T[12];
#pragma unroll
    for (int e = 0; e < 12; ++e) T[e] = 0.0f;
    const float* wv = w + (size_t)v * 16;
#pragma unroll 4
    for (int jj = 0; jj < 16; ++jj) {
        float ww = wv[jj];
        const float* aj = &As[jj * 12];
#pragma unroll
        for (int e = 0; e < 12; ++e) T[e] += ww * aj[e];
    }
    const float* pv = vp + (size_t)b * VCP + v * 3;
    float x = pv[0], y = pv[1], z = pv[2];
    float ox = (T[0]*x + T[1]*y + T[2]*z  + T[3])  * SCALE_;
    float oy = (T[4]*x + T[5]*y + T[6]*z  + T[7])  * SCALE_;
    float oz = (T[8]*x + T[9]*y + T[10]*z + T[11]) * SCALE_;
    float* vo = verts + ((size_t)b * V_ + v) * 3;
    vo[0] = ox; vo[1] = oy; vo[2] = oz;
    const int tips[5] = {333, 444, 672, 555, 744};
#pragma unroll
    for (int k = 0; k < 5; ++k) {
        if (v == tips[k]) {
            float* jo = joints + ((size_t)b * 21 + 16 + k) * 3;
            jo[0] = ox; jo[1] = oy; jo[2] = oz;
        }
    }
}

// ---------------------------------------------------------------------------
extern "C" void kernel_launch(void* const* d_in, const int* in_sizes, int n_in,
                              void* d_out, int out_size, void* d_ws, size_t ws_size,
                              hipStream_t stream)
{
    (void)in_sizes; (void)n_in; (void)out_size; (void)ws_size;
    const float* beta       = (const float*)d_in[0];
    const float* theta      = (const float*)d_in[1];
    const float* v_template = (const float*)d_in[2];
    const float* shapedirs  = (const float*)d_in[3];
    const float* posedirs   = (const float*)d_in[4];
    const float* Jreg       = (const float*)d_in[5];
    const float* weights    = (const float*)d_in[6];
    const float* hands_comp = (const float*)d_in[7];
    const float* hands_mean = (const float*)d_in[8];

    float* ws = (float*)d_ws;
    size_t o = 0;
    float* vs   = ws + o; o += (size_t)B_ * VCP;   // v_shaped -> v_posed (19.1 MB)
    float* pdp  = ws + o; o += (size_t)KP * VCP;   // padded posedirs
    float* pf   = ws + o; o += (size_t)B_ * KP;    // pose_feat padded
    float* Rs   = ws + o; o += (size_t)B_ * 144;   // rotations
    float* J    = ws + o; o += (size_t)B_ * 48;    // joints (rest)
    float* Amat = ws + o; o += (size_t)B_ * 192;   // skinning transforms
    // total ~6.17M floats (~24.7 MB) of workspace

    float* verts  = (float*)d_out;
    float* joints = verts + (size_t)B_ * V_ * 3;

    k_pad_posedirs<<<dim3((KP * VCP + 255) / 256), 256, 0, stream>>>(posedirs, pdp);
    k_shape<<<dim3((VCP + 255) / 256, B_), 256, 0, stream>>>(beta, shapedirs, v_template, vs);
    k_pose<<<dim3(B_), 64, 0, stream>>>(theta, hands_comp, hands_mean, Rs, pf);
    k_jreg<<<dim3(B_), 64, 0, stream>>>(vs, Jreg, J);             // before in-place GEMM
    k_kinematics<<<dim3(B_), 32, 0, stream>>>(Rs, J, Amat, joints);
    k_gemm_posedirs<<<dim3(VCP / 16, B_ / 16), 32, 0, stream>>>(pf, pdp, vs);
    k_skin<<<dim3((V_ + 255) / 256, B_), 256, 0, stream>>>(vs, Amat, weights, verts, joints);
}